// DualCobdryTransformer_11596411699256
// MI455X (gfx1250) — compile-verified
//
#include <hip/hip_runtime.h>
#include <math.h>

// Problem constants (match reference)
#define E_N 50000
#define F_N 25000
#define M_N 100000
#define P_N 75000
#define DD  128
#define H   3
#define DH  128
#define HD  384   // H * DH

typedef __attribute__((ext_vector_type(2))) float v2f;
typedef __attribute__((ext_vector_type(8))) float v8f;

// ---------------------------------------------------------------------------
// WMMA fp32 GEMM:  C[M x NT] = (gather/sign A)[M x 128] @ B[128 x NT] (+bias) (+=C)
// Block = 256 threads = 8 waves; block tile = 64 rows x 128 cols.
// NT (output row stride) is a template constant so all C/B addressing folds
// into immediate offsets. Each wave owns a 16-col strip with 4 row-subtiles
// (4 accumulators): every B fragment feeds 4 v_wmma_f32_16x16x4_f32.
// ---------------------------------------------------------------------------
template<int NT, bool HIDX, bool HSGN, bool HBIAS, bool ACC>
__global__ void gemm_wmma_k(const float* __restrict__ A,
                            const int* __restrict__ idx, int idxStride,
                            const int* __restrict__ sgn, int sgnStride,
                            const float* __restrict__ B,
                            const float* __restrict__ bias,
                            float* __restrict__ C, int M)
{
    __shared__ float As[64][132];   // 128 -> 132 float stride: spreads LDS banks
    __shared__ int   rmap[64];
    __shared__ float smap[64];

    const int tid = threadIdx.x;
    const int r0  = blockIdx.x * 64;

    // Stage row indices / signs once (instead of per A element).
    if (tid < 64) {
        int row = r0 + tid;
        int ar = -1; float sg = 1.0f;
        if (row < M) {
            ar = HIDX ? idx[(size_t)row * idxStride] : row;
            if (HSGN) sg = (float)sgn[(size_t)row * sgnStride];
        }
        rmap[tid] = ar;
        smap[tid] = sg;
    }
    __syncthreads();

    // Cooperative A-tile load: float4 (b128) loads/stores, 8 per thread.
    for (int i = tid; i < 64 * 32; i += 256) {
        int r  = i >> 5;
        int c4 = (i & 31) * 4;
        int ar = rmap[r];
        float4 v = make_float4(0.0f, 0.0f, 0.0f, 0.0f);
        if (ar >= 0) v = *(const float4*)&A[(size_t)ar * DD + c4];
        if (HSGN) {
            float sg = smap[r];
            v.x *= sg; v.y *= sg; v.z *= sg; v.w *= sg;
        }
        *(float4*)&As[r][c4] = v;   // row stride 132f = 528B, 16B-aligned
    }
    __syncthreads();

    const int wave = tid >> 5;
    const int lane = tid & 31;
    const int half = lane >> 4;     // 0: lanes 0-15, 1: lanes 16-31
    const int sub  = lane & 15;
    const int c0   = blockIdx.y * 128 + wave * 16;

    v8f acc0 = {}, acc1 = {}, acc2 = {}, acc3 = {};
    // Per-lane B base: row 2*half, column c0+sub; loop offsets are k0*NT (imm).
    const float* Bp = B + (size_t)(2 * half) * NT + c0 + sub;

    #pragma unroll 2
    for (int k0 = 0; k0 < 128; k0 += 4) {
        const int kk = k0 + 2 * half;
        v2f b;
        b.x = Bp[(size_t)k0 * NT];
        b.y = Bp[(size_t)(k0 + 1) * NT];
        v2f a0 = *(const v2f*)&As[ 0 + sub][kk];
        v2f a1 = *(const v2f*)&As[16 + sub][kk];
        v2f a2 = *(const v2f*)&As[32 + sub][kk];
        v2f a3 = *(const v2f*)&As[48 + sub][kk];
        acc0 = __builtin_amdgcn_wmma_f32_16x16x4_f32(false, a0, false, b, (short)0, acc0, false, false);
        acc1 = __builtin_amdgcn_wmma_f32_16x16x4_f32(false, a1, false, b, (short)0, acc1, false, false);
        acc2 = __builtin_amdgcn_wmma_f32_16x16x4_f32(false, a2, false, b, (short)0, acc2, false, false);
        acc3 = __builtin_amdgcn_wmma_f32_16x16x4_f32(false, a3, false, b, (short)0, acc3, false, false);
    }

    const int col = c0 + sub;
    const float bv = HBIAS ? bias[col] : 0.0f;
    v8f accs[4] = {acc0, acc1, acc2, acc3};

    // Base pointer for this lane's first output row; all further offsets are
    // compile-time multiples of NT -> immediate-offset global stores.
    float* Cp = C + (size_t)(r0 + half * 8) * NT + col;

    if (r0 + 64 <= M) {
        // Interior block: branch-free epilogue.
        #pragma unroll
        for (int rt = 0; rt < 4; ++rt) {
            #pragma unroll
            for (int i = 0; i < 8; ++i) {
                size_t o = (size_t)(rt * 16 + i) * NT;
                float r = accs[rt][i];
                if (HBIAS) r += bv;
                if (ACC)   r += Cp[o];
                Cp[o] = r;
            }
        }
    } else {
        // Boundary block (at most one per grid column): guarded.
        #pragma unroll
        for (int rt = 0; rt < 4; ++rt) {
            #pragma unroll
            for (int i = 0; i < 8; ++i) {
                int row = r0 + rt * 16 + half * 8 + i;
                if (row < M) {
                    size_t o = (size_t)(rt * 16 + i) * NT;
                    float r = accs[rt][i];
                    if (HBIAS) r += bv;
                    if (ACC)   r += Cp[o];
                    Cp[o] = r;
                }
            }
        }
    }
}

// ---------------------------------------------------------------------------
// h_aggr[f] += h_dE[e] * sgn  (one thread per (edge, dim))
// ---------------------------------------------------------------------------
__global__ void scatter_aggr_k(const float* __restrict__ hE,
                               const int* __restrict__ cob,
                               float* __restrict__ hagg)
{
    size_t t = (size_t)blockIdx.x * blockDim.x + threadIdx.x;
    if (t >= (size_t)M_N * DD) return;
    int m = (int)(t >> 7), d = (int)(t & 127);
    int e = cob[(size_t)m * 3 + 0];
    int f = cob[(size_t)m * 3 + 1];
    float sg = (float)cob[(size_t)m * 3 + 2];
    atomicAdd(&hagg[(size_t)f * DD + d], hE[(size_t)e * DD + d] * sg);
}

__global__ void fill_f32_k(float* __restrict__ p, float v, size_t n)
{
    size_t t = (size_t)blockIdx.x * blockDim.x + threadIdx.x;
    if (t < n) p[t] = v;
}

__device__ inline void atomicMaxFloat(float* addr, float val)
{
    if (val >= 0.0f) atomicMax((int*)addr, __float_as_int(val));
    else             atomicMin((unsigned int*)addr, __float_as_uint(val));
}

// ---------------------------------------------------------------------------
// Pass 1: alpha[m,h] = <q[dst,h,:], k[src,h,:]+e[m,h,:]> / sqrt(DH); segment max.
// One wave32 per (edge, head).
// ---------------------------------------------------------------------------
__global__ void attn_alpha_k(const float* __restrict__ q,
                             const float* __restrict__ k,
                             const float* __restrict__ e,
                             const int* __restrict__ srcIdx, int ss,
                             const int* __restrict__ dstIdx, int ds,
                             float* __restrict__ alpha,
                             float* __restrict__ amax, int P)
{
    int wid  = (int)(((size_t)blockIdx.x * blockDim.x + threadIdx.x) >> 5);
    int lane = threadIdx.x & 31;
    if (wid >= P * H) return;
    int m = wid / H, hh = wid - m * H;
    int src = srcIdx[(size_t)m * ss];
    int dst = dstIdx[(size_t)m * ds];
    const float* qr = q + (size_t)dst * HD + hh * DH;
    const float* kr = k + (size_t)src * HD + hh * DH;
    const float* er = e + (size_t)m   * HD + hh * DH;
    float sum = 0.0f;
    #pragma unroll
    for (int d = lane; d < DH; d += 32) sum += qr[d] * (kr[d] + er[d]);
    #pragma unroll
    for (int off = 16; off > 0; off >>= 1) sum += __shfl_xor(sum, off, 32);
    if (lane == 0) {
        float a = sum * 0.08838834764831845f;   // 1/sqrt(128)
        alpha[(size_t)m * H + hh] = a;
        atomicMaxFloat(&amax[(size_t)dst * H + hh], a);
    }
}

// Pass 2: ex = exp(alpha - amax[dst]); den[dst] += ex (one thread per (m,h))
__global__ void attn_expsum_k(float* __restrict__ alpha,
                              const float* __restrict__ amax,
                              float* __restrict__ den,
                              const int* __restrict__ dstIdx, int ds, int P)
{
    int t = (int)((size_t)blockIdx.x * blockDim.x + threadIdx.x);
    if (t >= P * H) return;
    int m = t / H, hh = t - m * H;
    int dst = dstIdx[(size_t)m * ds];
    float am = amax[(size_t)dst * H + hh];
    if (!isfinite(am)) am = 0.0f;
    float ex = __expf(alpha[t] - am);
    alpha[t] = ex;
    atomicAdd(&den[(size_t)dst * H + hh], ex);
}

// Pass 3: acc[dst,h,:] += (ex/den) * (v[src,h,:] + e[m,h,:])  (block per (m,h))
__global__ void attn_scatter_k(const float* __restrict__ v,
                               const float* __restrict__ e,
                               const float* __restrict__ alpha,
                               const float* __restrict__ den,
                               const int* __restrict__ srcIdx, int ss,
                               const int* __restrict__ dstIdx, int ds,
                               float* __restrict__ acc, int P)
{
    int b = blockIdx.x;
    int m = b / H, hh = b - m * H;
    if (m >= P) return;
    int d = threadIdx.x;
    int src = srcIdx[(size_t)m * ss];
    int dst = dstIdx[(size_t)m * ds];
    float w = alpha[(size_t)m * H + hh] / (den[(size_t)dst * H + hh] + 1e-16f);
    float val = v[(size_t)src * HD + hh * DH + d] + e[(size_t)m * HD + hh * DH + d];
    atomicAdd(&acc[((size_t)dst * H + hh) * DH + d], w * val);
}

// h[i] = mean_h(acc) + s[i] + h[i]  (residual, in place)
__global__ void finalize_res_k(float* __restrict__ h,
                               const float* __restrict__ acc,
                               const float* __restrict__ s, int n)
{
    size_t t = (size_t)blockIdx.x * blockDim.x + threadIdx.x;
    if (t >= (size_t)n * DD) return;
    int f = (int)(t >> 7), d = (int)(t & 127);
    float m = (acc[((size_t)f * H + 0) * DH + d] +
               acc[((size_t)f * H + 1) * DH + d] +
               acc[((size_t)f * H + 2) * DH + d]) * (1.0f / 3.0f);
    h[t] = m + s[t] + h[t];
}

// out[i] = mean_h(acc) + s[i]
__global__ void finalize_out_k(float* __restrict__ out,
                               const float* __restrict__ acc,
                               const float* __restrict__ s, int n)
{
    size_t t = (size_t)blockIdx.x * blockDim.x + threadIdx.x;
    if (t >= (size_t)n * DD) return;
    int f = (int)(t >> 7), d = (int)(t & 127);
    float m = (acc[((size_t)f * H + 0) * DH + d] +
               acc[((size_t)f * H + 1) * DH + d] +
               acc[((size_t)f * H + 2) * DH + d]) * (1.0f / 3.0f);
    out[t] = m + s[t];
}

// ---------------------------------------------------------------------------
// Host-side dispatch: pick the specialized GEMM instantiation.
// ---------------------------------------------------------------------------
static inline void launch_gemm(const float* A, const int* idx, int idxStride,
                               const int* sgn, int sgnStride,
                               const float* B, const float* bias, float* C,
                               int M, int N, int accum, hipStream_t s)
{
    dim3 grid((M + 63) / 64, N >> 7);
    if (N == 384) {
        if (idx) {
            if (sgn)
                gemm_wmma_k<384, true, true, false, false><<<grid, 256, 0, s>>>(
                    A, idx, idxStride, sgn, sgnStride, B, bias, C, M);
            else
                gemm_wmma_k<384, true, false, false, false><<<grid, 256, 0, s>>>(
                    A, idx, idxStride, sgn, sgnStride, B, bias, C, M);
        } else {
            gemm_wmma_k<384, false, false, true, false><<<grid, 256, 0, s>>>(
                A, idx, idxStride, sgn, sgnStride, B, bias, C, M);
        }
    } else { // N == 128
        if (accum)
            gemm_wmma_k<128, false, false, true, true><<<grid, 256, 0, s>>>(
                A, idx, idxStride, sgn, sgnStride, B, bias, C, M);
        else
            gemm_wmma_k<128, false, false, true, false><<<grid, 256, 0, s>>>(
                A, idx, idxStride, sgn, sgnStride, B, bias, C, M);
    }
}

extern "C" void kernel_launch(void* const* d_in, const int* in_sizes, int n_in,
                              void* d_out, int out_size, void* d_ws, size_t ws_size,
                              hipStream_t stream)
{
    const float* h_dE   = (const float*)d_in[0];
    const float* h_dF   = (const float*)d_in[1];
    const int*   cob    = (const int*)  d_in[2];
    const int*   nn     = (const int*)  d_in[3];
    const float* aggr_W = (const float*)d_in[4];
    const float* aggr_b = (const float*)d_in[5];
    const float* ctr_W  = (const float*)d_in[6];
    const float* ctr_b  = (const float*)d_in[7];
    const float* f_Wq   = (const float*)d_in[8];
    const float* f_bq   = (const float*)d_in[9];
    const float* f_Wk   = (const float*)d_in[10];
    const float* f_bk   = (const float*)d_in[11];
    const float* f_Wv   = (const float*)d_in[12];
    const float* f_bv   = (const float*)d_in[13];
    const float* f_We   = (const float*)d_in[14];
    const float* f_Ws   = (const float*)d_in[15];
    const float* f_bs   = (const float*)d_in[16];
    const float* e_Wq   = (const float*)d_in[17];
    const float* e_bq   = (const float*)d_in[18];
    const float* e_Wk   = (const float*)d_in[19];
    const float* e_bk   = (const float*)d_in[20];
    const float* e_Wv   = (const float*)d_in[21];
    const float* e_bv   = (const float*)d_in[22];
    const float* e_We   = (const float*)d_in[23];
    const float* e_Ws   = (const float*)d_in[24];
    const float* e_bs   = (const float*)d_in[25];

    // Workspace layout (floats)
    float* ws = (float*)d_ws;
    size_t off = 0;
    float* h     = ws + off; off += (size_t)F_N * DD;
    float* hagg  = ws + off; off += (size_t)F_N * DD;
    float* qb    = ws + off; off += (size_t)E_N * HD;     // max(E_N,F_N) rows
    float* kb    = ws + off; off += (size_t)F_N * HD;
    float* vb    = ws + off; off += (size_t)F_N * HD;
    float* eb    = ws + off; off += (size_t)M_N * HD;     // max(P_N,M_N) rows
    float* sb    = ws + off; off += (size_t)E_N * DD;
    float* alpha = ws + off; off += (size_t)M_N * H;
    float* amax  = ws + off; off += (size_t)E_N * H;
    float* den   = ws + off; off += (size_t)E_N * H;
    float* acc   = ws + off; off += (size_t)E_N * H * DH;

    // ---- Stage 1: h = (segsum h_dE*sgn -> f) @ aggr_W + aggr_b + h_dF @ ctr_W + ctr_b
    hipMemsetAsync(hagg, 0, (size_t)F_N * DD * sizeof(float), stream);
    {
        size_t tot = (size_t)M_N * DD;
        scatter_aggr_k<<<(unsigned)((tot + 255) / 256), 256, 0, stream>>>(h_dE, cob, hagg);
    }
    launch_gemm(hagg, nullptr, 0, nullptr, 0, aggr_W, aggr_b, h, F_N, DD, 0, stream);
    launch_gemm(h_dF, nullptr, 0, nullptr, 0, ctr_W,  ctr_b,  h, F_N, DD, 1, stream);

    // ---- Stage 2: three transformer-conv layers on F nodes (edges = nn_idx)
    const int* nn_src = nn + 0;   // stride 3
    const int* nn_dst = nn + 1;
    const int* nn_ea  = nn + 2;
    for (int l = 0; l < 3; ++l) {
        const size_t wofs = (size_t)l * DD * HD;
        launch_gemm(h, nullptr, 0, nullptr, 0, f_Wq + wofs, f_bq + (size_t)l * HD, qb, F_N, HD, 0, stream);
        launch_gemm(h, nullptr, 0, nullptr, 0, f_Wk + wofs, f_bk + (size_t)l * HD, kb, F_N, HD, 0, stream);
        launch_gemm(h, nullptr, 0, nullptr, 0, f_Wv + wofs, f_bv + (size_t)l * HD, vb, F_N, HD, 0, stream);
        launch_gemm(h, nullptr, 0, nullptr, 0, f_Ws + (size_t)l * DD * DD, f_bs + (size_t)l * DD, sb, F_N, DD, 0, stream);
        // e = h_dE[nn[:,2]] @ f_We[l]   (gather fused into GEMM A-load)
        launch_gemm(h_dE, nn_ea, 3, nullptr, 0, f_We + wofs, nullptr, eb, P_N, HD, 0, stream);

        fill_f32_k<<<(unsigned)(((size_t)F_N * H + 255) / 256), 256, 0, stream>>>(amax, -INFINITY, (size_t)F_N * H);
        hipMemsetAsync(den, 0, (size_t)F_N * H * sizeof(float), stream);
        hipMemsetAsync(acc, 0, (size_t)F_N * H * DH * sizeof(float), stream);

        {
            size_t waves = (size_t)P_N * H;
            attn_alpha_k<<<(unsigned)((waves * 32 + 255) / 256), 256, 0, stream>>>(
                qb, kb, eb, nn_src, 3, nn_dst, 3, alpha, amax, P_N);
            attn_expsum_k<<<(unsigned)((waves + 255) / 256), 256, 0, stream>>>(
                alpha, amax, den, nn_dst, 3, P_N);
            attn_scatter_k<<<(unsigned)waves, DH, 0, stream>>>(
                vb, eb, alpha, den, nn_src, 3, nn_dst, 3, acc, P_N);
        }
        finalize_res_k<<<(unsigned)(((size_t)F_N * DD + 255) / 256), 256, 0, stream>>>(h, acc, sb, F_N);
    }

    // ---- Stage 3: final tconv F -> E (edges = cobdry; src=f_col, dst=e_col)
    const int* cb_e  = cob + 0;   // dst (E side), stride 3
    const int* cb_f  = cob + 1;   // src (F side)
    const int* cb_sg = cob + 2;
    launch_gemm(h_dE, nullptr, 0, nullptr, 0, e_Wq, e_bq, qb, E_N, HD, 0, stream);
    launch_gemm(h,    nullptr, 0, nullptr, 0, e_Wk, e_bk, kb, F_N, HD, 0, stream);
    launch_gemm(h,    nullptr, 0, nullptr, 0, e_Wv, e_bv, vb, F_N, HD, 0, stream);
    launch_gemm(h_dE, nullptr, 0, nullptr, 0, e_Ws, e_bs, sb, E_N, DD, 0, stream);
    // eattr = (h[f_col] * sgn) @ e_We  (gather + sign fused into GEMM A-load)
    launch_gemm(h, cb_f, 3, cb_sg, 3, e_We, nullptr, eb, M_N, HD, 0, stream);

    fill_f32_k<<<(unsigned)(((size_t)E_N * H + 255) / 256), 256, 0, stream>>>(amax, -INFINITY, (size_t)E_N * H);
    hipMemsetAsync(den, 0, (size_t)E_N * H * sizeof(float), stream);
    hipMemsetAsync(acc, 0, (size_t)E_N * H * DH * sizeof(float), stream);

    {
        size_t waves = (size_t)M_N * H;
        attn_alpha_k<<<(unsigned)((waves * 32 + 255) / 256), 256, 0, stream>>>(
            qb, kb, eb, cb_f, 3, cb_e, 3, alpha, amax, M_N);
        attn_expsum_k<<<(unsigned)((waves + 255) / 256), 256, 0, stream>>>(
            alpha, amax, den, cb_e, 3, M_N);
        attn_scatter_k<<<(unsigned)waves, DH, 0, stream>>>(
            vb, eb, alpha, den, cb_f, 3, cb_e, 3, acc, M_N);
    }
    finalize_out_k<<<(unsigned)(((size_t)E_N * DD + 255) / 256), 256, 0, stream>>>(
        (float*)d_out, acc, sb, E_N);
}